// DynamicSpatialFilter_28741921145061
// MI455X (gfx1250) — compile-verified
//
#include <hip/hip_runtime.h>
#include <hip/hip_bf16.h>

typedef __attribute__((ext_vector_type(2))) float v2f;
typedef __attribute__((ext_vector_type(8))) float v8f;

#define N_CH   64
#define T_LEN  4000
#define N_MLP  4160   // 64*(64+1)

// ---------------- Stage 1: feats[b,c] = log(var_unbiased(x[b,c,:])) ----------------
// One wave (32 lanes) per (b,c) row; float4 coalesced streaming.
__global__ __launch_bounds__(256) void dsf_var_kernel(const float* __restrict__ x,
                                                      float* __restrict__ feats) {
    const int gw   = blockIdx.x * 8 + (threadIdx.x >> 5);   // 0..16383 rows
    const int lane = threadIdx.x & 31;
    const float4* row4 = (const float4*)(x + (size_t)gw * T_LEN);

    float s = 0.f, ss = 0.f;
    for (int i = lane; i < T_LEN / 4; i += 32) {
        float4 v = row4[i];
        s  += v.x + v.y + v.z + v.w;
        ss += v.x * v.x + v.y * v.y + v.z * v.z + v.w * v.w;
    }
    // wave32 reduction
    for (int off = 16; off > 0; off >>= 1) {
        s  += __shfl_xor(s,  off, 32);
        ss += __shfl_xor(ss, off, 32);
    }
    if (lane == 0) {
        const float n = (float)T_LEN;
        float var = (ss - s * s / n) / (n - 1.f);
        float lg  = (var > 0.f) ? __logf(var) : 0.f;   // -inf -> 0 per reference
        feats[gw] = lg;
    }
}

// ---------------- Stage 2: per-batch hypernetwork MLP ----------------
// h = relu(feats@W1+b1); mlp = h@W2+b2; split into bias (first 64) and W (64x64).
__global__ __launch_bounds__(256) void dsf_mlp_kernel(const float* __restrict__ feats,
                                                      const float* __restrict__ W1,
                                                      const float* __restrict__ b1,
                                                      const float* __restrict__ W2,
                                                      const float* __restrict__ b2,
                                                      float* __restrict__ Wws,
                                                      float* __restrict__ biasWs) {
    __shared__ float f[N_CH];
    __shared__ float h[N_CH];
    const int b = blockIdx.x;
    const int tid = threadIdx.x;

    if (tid < N_CH) f[tid] = feats[b * N_CH + tid];
    __syncthreads();
    if (tid < N_CH) {
        float s = b1[tid];
        for (int c = 0; c < N_CH; ++c) s += f[c] * W1[c * N_CH + tid];
        h[tid] = fmaxf(s, 0.f);
    }
    __syncthreads();
    for (int j = tid; j < N_MLP; j += 256) {
        float s = b2[j];
        for (int k = 0; k < N_CH; ++k) s += h[k] * W2[k * N_MLP + j];
        if (j < 64) biasWs[b * 64 + j] = s;
        else        Wws[(size_t)b * 4096 + (j - 64)] = s;   // row-major (o, c)
    }
}

// ---------------- Stage 3: out[b] = W[b] @ x[b] + bias[b]  (fp32 WMMA) ----------------
// Block: 8 waves; each wave owns a 16-column slab; grid.y * 128 covers T=4000.
__global__ __launch_bounds__(256) void dsf_matmul_kernel(const float* __restrict__ x,
                                                         const float* __restrict__ Wws,
                                                         const float* __restrict__ biasWs,
                                                         float* __restrict__ out) {
    __shared__ float Wlds[64 * 65];   // row-pad 65 floats: kills 16-way bank conflicts
    __shared__ float blds[64];

    const int b   = blockIdx.x;
    const int tid = threadIdx.x;

    // stage W[b] (4096 floats) + bias[b] into LDS
    const float* Wsrc = Wws + (size_t)b * 4096;
    for (int i = tid; i < 4096; i += 256) {
        int r = i >> 6, c = i & 63;
        Wlds[r * 65 + c] = Wsrc[i];
    }
    if (tid < 64) blds[tid] = biasWs[b * 64 + tid];
    __syncthreads();

    const int wave = tid >> 5;
    const int lane = tid & 31;
    const int n0   = blockIdx.y * 128 + wave * 16;   // column slab base
    if (n0 >= T_LEN) return;                          // uniform per wave (4000 % 16 == 0)

    const int half = lane >> 4;     // 0: lanes 0-15, 1: lanes 16-31
    const int l16  = lane & 15;

    const float* xb = x + (size_t)b * N_CH * T_LEN;
    v8f acc0 = {}, acc1 = {}, acc2 = {}, acc3 = {};

    for (int k = 0; k < 16; ++k) {                    // K = 64 in steps of 4
        // B-frag (4x16): lane holds (K = vgpr + 2*half, N = n0 + l16)
        const int krow = k * 4 + 2 * half;
        v2f bf;
        bf.x = xb[(size_t)krow       * T_LEN + n0 + l16];
        bf.y = xb[(size_t)(krow + 1) * T_LEN + n0 + l16];

        // A-frags (16x4) from LDS: lane holds (M = mtile*16 + l16, K = vgpr + 2*half)
        #pragma unroll
        for (int m = 0; m < 4; ++m) {
            const float* wp = &Wlds[(m * 16 + l16) * 65 + k * 4 + 2 * half];
            v2f af; af.x = wp[0]; af.y = wp[1];
            v8f* cc = (m == 0) ? &acc0 : (m == 1) ? &acc1 : (m == 2) ? &acc2 : &acc3;
            *cc = __builtin_amdgcn_wmma_f32_16x16x4_f32(
                /*neg_a=*/false, af, /*neg_b=*/false, bf,
                /*c_mod=*/(short)0, *cc, /*reuse_a=*/false, /*reuse_b=*/false);
        }
    }

    // D layout: vgpr r -> M = mtile*16 + r + 8*half, N = n0 + l16
    float* ob = out + (size_t)b * N_CH * T_LEN;
    #pragma unroll
    for (int m = 0; m < 4; ++m) {
        const v8f& cc = (m == 0) ? acc0 : (m == 1) ? acc1 : (m == 2) ? acc2 : acc3;
        #pragma unroll
        for (int r = 0; r < 8; ++r) {
            const int M = m * 16 + r + 8 * half;
            ob[(size_t)M * T_LEN + n0 + l16] = cc[r] + blds[M];
        }
    }
}

extern "C" void kernel_launch(void* const* d_in, const int* in_sizes, int n_in,
                              void* d_out, int out_size, void* d_ws, size_t ws_size,
                              hipStream_t stream) {
    const float* x  = (const float*)d_in[0];   // 256*1*64*4000
    const float* W1 = (const float*)d_in[1];   // 64*64
    const float* b1 = (const float*)d_in[2];   // 64
    const float* W2 = (const float*)d_in[3];   // 64*4160
    const float* b2 = (const float*)d_in[4];   // 4160
    float* out = (float*)d_out;                // 256*64*4000

    // workspace layout
    float* feats  = (float*)d_ws;              // 16384 floats
    float* biasWs = feats + 256 * 64;          // 16384 floats
    float* Wws    = biasWs + 256 * 64;         // 256*4096 floats  (~4.3 MB total)

    dsf_var_kernel<<<2048, 256, 0, stream>>>(x, feats);
    dsf_mlp_kernel<<<256, 256, 0, stream>>>(feats, W1, b1, W2, b2, Wws, biasWs);
    dsf_matmul_kernel<<<dim3(256, 32), 256, 0, stream>>>(x, Wws, biasWs, out);
}